// PartitionedCliffordAlgebra_69492570849780
// MI455X (gfx1250) — compile-verified
//
#include <hip/hip_runtime.h>

// Cl(4,2,0): N=6 basis vectors, DIM=64 blades, BATCH=32768.
// out[n,k] = sum_{a,b: a^b=k} sign(a,b) * x[n,a] * y[n,b]
// Signs are compile-time; template recursion forces full unrolling so all
// arrays live in VGPRs (no scratch spills).

#define DIM 64
#define ROWS_PER_BLOCK 32   // one wave32 per block, one row per lane
#define KHALF 32            // outputs computed per pass (register pressure control)

// ---- compile-time Cayley sign, faithful to the reference ----
__host__ __device__ constexpr float cayley_sign(int a, int b) {
    int swaps = 0;
    for (int bit = 0; bit < 6; ++bit) {
        if ((a >> bit) & 1) {
            swaps += __builtin_popcount((unsigned)(b & ((1 << bit) - 1)));
        }
    }
    // negative-signature bits [p, p+q) = bits [4,6) -> mask 0b110000
    swaps += __builtin_popcount((unsigned)(a & b & 0x30));
    return (swaps & 1) ? -1.0f : 1.0f;
}

// ---- forced-unroll MAC engine: all indices are template constants ----
template<int A, int J, int KBASE>
struct MacB {
    __device__ static __attribute__((always_inline))
    void run(float (&acc)[KHALF], const float (&yv)[DIM], float xa) {
        constexpr int   k = KBASE + J;
        constexpr int   b = A ^ k;
        constexpr float s = cayley_sign(A, b);
        acc[J] = fmaf(s * xa, yv[b], acc[J]);   // +-1 folds into fma neg modifier
        MacB<A, J + 1, KBASE>::run(acc, yv, xa);
    }
};
template<int A, int KBASE>
struct MacB<A, KHALF, KBASE> {
    __device__ static __attribute__((always_inline))
    void run(float (&)[KHALF], const float (&)[DIM], float) {}
};

template<int A0, int KBASE>
struct MacChunk {
    __device__ static __attribute__((always_inline))
    void run(float (&acc)[KHALF], const float (&yv)[DIM], const float4* xr4) {
        float4 xv = xr4[A0 / 4];                 // ds_load_b128, 4 x-values live
        MacB<A0 + 0, 0, KBASE>::run(acc, yv, xv.x);
        MacB<A0 + 1, 0, KBASE>::run(acc, yv, xv.y);
        MacB<A0 + 2, 0, KBASE>::run(acc, yv, xv.z);
        MacB<A0 + 3, 0, KBASE>::run(acc, yv, xv.w);
        MacChunk<A0 + 4, KBASE>::run(acc, yv, xr4);
    }
};
template<int KBASE>
struct MacChunk<DIM, KBASE> {
    __device__ static __attribute__((always_inline))
    void run(float (&)[KHALF], const float (&)[DIM], const float4*) {}
};

// ---- ext-vector types for the gfx1250 TDM builtin (6-arg form) ----
typedef unsigned int v4u __attribute__((ext_vector_type(4)));
typedef int          v8i __attribute__((ext_vector_type(8)));
typedef int          v4i __attribute__((ext_vector_type(4)));

__global__ __launch_bounds__(ROWS_PER_BLOCK, 1)
void clifford_geoprod_kernel(const float* __restrict__ x,
                             const float* __restrict__ y,
                             float* __restrict__ out,
                             int batch)
{
    __shared__ __align__(16) float smem[2 * ROWS_PER_BLOCK * DIM]; // 16 KB

    const int tid  = threadIdx.x;
    const int row0 = blockIdx.x * ROWS_PER_BLOCK;
    if (row0 >= batch) return;   // batch=32768 divides evenly; guard anyway

    const unsigned tile_elems = ROWS_PER_BLOCK * DIM;              // 2048 f32
    unsigned long long gx = (unsigned long long)(x + (size_t)row0 * DIM);
    unsigned long long gy = (unsigned long long)(y + (size_t)row0 * DIM);
    // flat LDS address: low 32 bits == LDS byte offset (aperture in high bits)
    unsigned lds_base = (unsigned)(unsigned long long)(void*)&smem[0];

    // ---- Tensor DMA Descriptor (D#), per CDNA5 ISA ----
    // Group 0: [1:0]=count=1 | [63:32]=lds_addr | [120:64]=global_addr | [127:126]=type=2
    v4u g0x = { 1u, lds_base,
                (unsigned)gx,
                (((unsigned)(gx >> 32)) & 0x01FFFFFFu) | 0x80000000u };
    v4u g0y = { 1u, lds_base + tile_elems * 4u,
                (unsigned)gy,
                (((unsigned)(gy >> 32)) & 0x01FFFFFFu) | 0x80000000u };
    // Group 1: data_size=4B (code 2); 1-D tile: tile_dim0 = tensor_dim0 = 2048,
    // tensor_dim1=1, tile_dim1/2=0 (unused), dim0 stride = 2048, no padding.
    v8i g1 = {
        (int)(2u << 16),                          // w0: data_size=2 (4B)
        (int)((tile_elems & 0xFFFFu) << 16),      // w1: tensor_dim0[15:0]
        (int)((tile_elems >> 16) | (1u << 16)),   // w2: tensor_dim0[31:16] | tensor_dim1=1
        (int)((tile_elems & 0xFFFFu) << 16),      // w3: tile_dim0 = 2048
        0,                                        // w4: tile_dim1=0, tile_dim2=0
        (int)tile_elems,                          // w5: tensor_dim0_stride[31:0]
        0, 0                                      // w6, w7
    };
    v4i gz4 = { 0, 0, 0, 0 };
    v8i gz8 = { 0, 0, 0, 0, 0, 0, 0, 0 };

    // One wave per block issues the DMA and is also the consumer: its own
    // TENSORcnt ordering suffices, no cross-wave barrier needed.
    __builtin_amdgcn_tensor_load_to_lds(g0x, g1, gz4, gz4, gz8, 0);  // X tile -> first 8 KB of smem in LDS
    __builtin_amdgcn_tensor_load_to_lds(g0y, g1, gz4, gz4, gz8, 0);  // Y tile -> second 8 KB of smem in LDS
    __builtin_amdgcn_s_wait_tensorcnt(0);
    asm volatile("" ::: "memory");   // keep ds_loads below the tensorcnt wait

    // ---- per-lane row pointers in LDS (256 B per row) ----
    const float4* xr4 = (const float4*)(smem + tid * DIM);
    const float4* yr4 = (const float4*)(smem + tile_elems + tid * DIM);

    // y row resident in registers (16x ds_load_b128)
    float yv[DIM];
#pragma unroll
    for (int j = 0; j < DIM / 4; ++j) {
        float4 v = yr4[j];
        yv[4 * j + 0] = v.x; yv[4 * j + 1] = v.y;
        yv[4 * j + 2] = v.z; yv[4 * j + 3] = v.w;
    }

    float4* op = (float4*)(out + (size_t)(row0 + tid) * DIM);

    // ---- pass 0: k in [0,32) -- 2048 fully-unrolled FMAs ----
    {
        float acc[KHALF];
#pragma unroll
        for (int j = 0; j < KHALF; ++j) acc[j] = 0.0f;
        MacChunk<0, 0>::run(acc, yv, xr4);
#pragma unroll
        for (int j = 0; j < KHALF / 4; ++j) {
            float4 v = { acc[4 * j + 0], acc[4 * j + 1],
                         acc[4 * j + 2], acc[4 * j + 3] };
            op[j] = v;                            // global_store_b128
        }
    }
    // ---- pass 1: k in [32,64) -- 2048 fully-unrolled FMAs ----
    {
        float acc[KHALF];
#pragma unroll
        for (int j = 0; j < KHALF; ++j) acc[j] = 0.0f;
        MacChunk<0, KHALF>::run(acc, yv, xr4);
#pragma unroll
        for (int j = 0; j < KHALF / 4; ++j) {
            float4 v = { acc[4 * j + 0], acc[4 * j + 1],
                         acc[4 * j + 2], acc[4 * j + 3] };
            op[KHALF / 4 + j] = v;
        }
    }
}

extern "C" void kernel_launch(void* const* d_in, const int* in_sizes, int n_in,
                              void* d_out, int out_size, void* d_ws, size_t ws_size,
                              hipStream_t stream) {
    const float* x = (const float*)d_in[0];
    const float* y = (const float*)d_in[1];
    // d_in[2] is the dense Cayley tensor; signs are baked in at compile time.
    float* out = (float*)d_out;

    const int batch  = in_sizes[0] / DIM;                             // 32768
    const int blocks = (batch + ROWS_PER_BLOCK - 1) / ROWS_PER_BLOCK; // 1024

    clifford_geoprod_kernel<<<blocks, ROWS_PER_BLOCK, 0, stream>>>(x, y, out, batch);
}